// GenerateReprojections_5987184411044
// MI455X (gfx1250) — compile-verified
//
#include <hip/hip_runtime.h>

typedef __attribute__((ext_vector_type(2))) float v2f;
typedef __attribute__((ext_vector_type(8))) float v8f;

#define V_ 2
#define B_ 16
#define C_ 3
#define H_ 320
#define W_ 1024

// ---------------------------------------------------------------------------
// Per-(view,batch) combined transform: out3x4 = (src_intr @ pose[:3,:3] @ Kinv | src_intr @ pose[:3,3])
// so that  src = out3x4 @ [d*x, d*y, d, 1]^T   reproduces the reference chain.
// ---------------------------------------------------------------------------
__global__ __launch_bounds__(32) void precompute_mats(
    const float* __restrict__ poses,     // [V,B,4,4]
    const float* __restrict__ tgt_intr,  // [B,4,4]
    const float* __restrict__ src_intr,  // [V,B,3,3]
    float* __restrict__ mats)            // [V,B,12] row-major 3x4
{
    int idx = threadIdx.x;
    if (idx >= V_ * B_) return;
    int v = idx / B_, b = idx % B_;

    // Kinv = inverse of tgt_intr[b][:3,:3] (block-triangular 4x4 => top-left inverse)
    const float* K = tgt_intr + b * 16;
    float k00 = K[0], k01 = K[1], k02 = K[2];
    float k10 = K[4], k11 = K[5], k12 = K[6];
    float k20 = K[8], k21 = K[9], k22 = K[10];
    float det = k00 * (k11 * k22 - k12 * k21)
              - k01 * (k10 * k22 - k12 * k20)
              + k02 * (k10 * k21 - k11 * k20);
    float id = 1.0f / det;
    float Ki[3][3] = {
        {(k11*k22 - k12*k21)*id, (k02*k21 - k01*k22)*id, (k01*k12 - k02*k11)*id},
        {(k12*k20 - k10*k22)*id, (k00*k22 - k02*k20)*id, (k02*k10 - k00*k12)*id},
        {(k10*k21 - k11*k20)*id, (k01*k20 - k00*k21)*id, (k00*k11 - k01*k10)*id}};

    // proj = src_intr[v,b] (3x3) @ poses[v,b][:3,:4]
    const float* S = src_intr + (size_t)(v * B_ + b) * 9;
    const float* P = poses    + (size_t)(v * B_ + b) * 16;
    float proj[3][4];
    #pragma unroll
    for (int r = 0; r < 3; ++r)
        #pragma unroll
        for (int c = 0; c < 4; ++c)
            proj[r][c] = S[r*3+0]*P[0*4+c] + S[r*3+1]*P[1*4+c] + S[r*3+2]*P[2*4+c];

    // [ proj[:,:3] @ Kinv | proj[:,3] ]
    float* out = mats + (size_t)idx * 12;
    #pragma unroll
    for (int r = 0; r < 3; ++r) {
        #pragma unroll
        for (int c = 0; c < 3; ++c)
            out[r*4+c] = proj[r][0]*Ki[0][c] + proj[r][1]*Ki[1][c] + proj[r][2]*Ki[2][c];
        out[r*4+3] = proj[r][3];
    }
}

// ---------------------------------------------------------------------------
// Main kernel: one block per (row y, batch b). 8 waves x 8 iterations x 16 px.
// Each wave does the projective transform for 16 pixels x 2 views with ONE
// V_WMMA_F32_16X16X4_F32, then every lane bilinear-gathers its (view,pixel).
// A-matrix: rows 0-2 = view0 [M|t], rows 8-10 = view1 (lands in lanes 16-31
// of the D accumulators). B-matrix column n = [d*x, d*y, d, 1].
// ---------------------------------------------------------------------------
__global__ __launch_bounds__(256) void reproject_kernel(
    const float* __restrict__ src_images, // [V,B,C,H,W]
    const float* __restrict__ depths,     // [B,1,H,W]
    const float* __restrict__ mats,       // [V,B,12]
    float* __restrict__ out)              // [V,B,C,H,W]
{
    const int y    = blockIdx.x;          // 0..H-1
    const int b    = blockIdx.y;          // 0..B-1
    const int tid  = (int)threadIdx.x;
    const int wave = tid >> 5;            // 0..7
    const int lane = tid & 31;
    const int n    = lane & 15;           // pixel slot / A-matrix row index
    const bool hi  = lane >= 16;          // K-half selector == view selector

    // ---- A operand (constant across the inner loop, 8x reuse) ----
    // lane<16 holds A[row=n][K=0..1], lane>=16 holds A[row=n][K=2..3].
    const int  vsel  = (n >= 8) ? 1 : 0;
    const int  r     = n - (vsel << 3);
    const bool valid = (r < 3);
    const float* mrow = mats + (size_t)(vsel * B_ + b) * 12 + r * 4 + (hi ? 2 : 0);
    v2f a;
    a.x = valid ? mrow[0] : 0.0f;
    a.y = valid ? mrow[1] : 0.0f;

    const float fy = (float)y;
    const int view = hi ? 1 : 0;
    const size_t plane = (size_t)H_ * W_;
    const float* img = src_images + (size_t)(view * B_ + b) * C_ * plane;
    float*       dvb = out        + (size_t)(view * B_ + b) * C_ * plane + (size_t)y * W_;

    // Folded normalize->denormalize constants:
    //   ix = px * (W/(W-1)) - 0.5 ;  iy = py * (H/(H-1)) - 0.5
    const float sclx = (float)W_ / (float)(W_ - 1);
    const float scly = (float)H_ / (float)(H_ - 1);

    #pragma unroll
    for (int g = 0; g < 8; ++g) {
        const int x  = (((g << 3) + wave) << 4) + n;   // pixel column
        const float d = depths[((size_t)b * H_ + y) * W_ + x];

        // ---- B operand ----
        v2f bb;
        bb.x = hi ? d    : d * (float)x;   // K=2 : K=0
        bb.y = hi ? 1.0f : d * fy;         // K=3 : K=1

        v8f c = {};
        c = __builtin_amdgcn_wmma_f32_16x16x4_f32(
                /*neg_a=*/false, a, /*neg_b=*/false, bb,
                /*c_mod=*/(short)0, c, /*reuse_a=*/false, /*reuse_b=*/false);

        // lane L: D rows 0/1/2 (view0) or 8/9/10 (view1), column n.
        const float sx = c[0], sy = c[1], sz = c[2];

        // ---- sample coordinates (1-ulp native rcp instead of IEEE divide) ----
        const float inv = __builtin_amdgcn_rcpf(sz + 1e-7f);
        float ix = fmaf(sx * inv, sclx, -0.5f);
        float iy = fmaf(sy * inv, scly, -0.5f);
        ix = fminf(fmaxf(ix, 0.0f), (float)(W_ - 1));
        iy = fminf(fmaxf(iy, 0.0f), (float)(H_ - 1));

        const float x0f = floorf(ix), y0f = floorf(iy);
        const float wx = ix - x0f, wy = iy - y0f;
        const int x0i = min(max((int)x0f, 0), W_ - 1);
        const int y0i = min(max((int)y0f, 0), H_ - 1);
        const int x1i = min(x0i + 1, W_ - 1);
        const int y1i = min(y0i + 1, H_ - 1);

        const int i00 = y0i * W_ + x0i, i01 = y0i * W_ + x1i;
        const int i10 = y1i * W_ + x0i, i11 = y1i * W_ + x1i;

        // ---- bilinear gather, 3 channels; writes are 16-wide contiguous ----
        #pragma unroll
        for (int ch = 0; ch < C_; ++ch) {
            const float* ip = img + (size_t)ch * plane;
            const float v00 = ip[i00], v01 = ip[i01];
            const float v10 = ip[i10], v11 = ip[i11];
            const float top = v00 + (v01 - v00) * wx;
            const float bot = v10 + (v11 - v10) * wx;
            dvb[(size_t)ch * plane + x] = top + (bot - top) * wy;
        }
    }
}

extern "C" void kernel_launch(void* const* d_in, const int* in_sizes, int n_in,
                              void* d_out, int out_size, void* d_ws, size_t ws_size,
                              hipStream_t stream) {
    const float* src_images = (const float*)d_in[0]; // [V,B,C,H,W]
    const float* depths     = (const float*)d_in[1]; // [B,1,H,W]
    const float* poses      = (const float*)d_in[2]; // [V,B,4,4]
    const float* tgt_intr   = (const float*)d_in[3]; // [B,4,4]
    const float* src_intr   = (const float*)d_in[4]; // [V,B,3,3]
    float* mats = (float*)d_ws;                      // V*B*12 floats = 1.5 KB

    precompute_mats<<<1, 32, 0, stream>>>(poses, tgt_intr, src_intr, mats);

    dim3 grid(H_, B_);
    reproject_kernel<<<grid, 256, 0, stream>>>(src_images, depths, mats, (float*)d_out);
}